// gcndec_68238440399151
// MI455X (gfx1250) — compile-verified
//
#include <hip/hip_runtime.h>
#include <hip/hip_bf16.h>

typedef __attribute__((ext_vector_type(2))) float v2f;
typedef __attribute__((ext_vector_type(8))) float v8f;

__device__ __forceinline__ float gelu_tanh(float x) {
    // jax.nn.gelu default (approximate=True): 0.5x(1+tanh(sqrt(2/pi)(x+0.044715x^3)))
    const float k0 = 0.7978845608028654f;
    const float k1 = 0.044715f;
    float x3 = x * x * x;
    return 0.5f * x * (1.0f + tanhf(k0 * (x + k1 * x3)));
}

__global__ void k_fill1(float* __restrict__ p, int n) {
    int i = blockIdx.x * blockDim.x + threadIdx.x;
    if (i < n) p[i] = 1.0f;
}

__global__ void k_deg(const long long* __restrict__ dst, float* __restrict__ deg, int E) {
    int e = blockIdx.x * blockDim.x + threadIdx.x;
    if (e < E) atomicAdd(&deg[(int)dst[e]], 1.0f);
}

__global__ void k_dinv(float* __restrict__ p, int n) {
    int i = blockIdx.x * blockDim.x + threadIdx.x;
    if (i < n) p[i] = rsqrtf(p[i]);   // deg >= 1 always (self loop)
}

__global__ void k_norm(const long long* __restrict__ src, const long long* __restrict__ dst,
                       const float* __restrict__ dinv, float* __restrict__ norm, int E) {
    int e = blockIdx.x * blockDim.x + threadIdx.x;
    if (e < E) norm[e] = dinv[(int)src[e]] * dinv[(int)dst[e]];
}

// Y[N,M] = X[N,K] @ W[K,M] (+ tvec[n]*wrow[m]) (+ bias[m]) (gelu optional)
// Block = 256 threads = 8 waves; all waves share one 64-column W panel, staged
// in LDS once per block. Each wave computes a 16x64 tile with 4x
// V_WMMA_F32_16X16X4_F32 accumulators; B operands come from ds_load, the only
// global load in the K-loop is the 8-byte A fetch.
__global__ __launch_bounds__(256) void k_linear_wmma(
    const float* __restrict__ X, const float* __restrict__ W,
    const float* __restrict__ bias, const float* __restrict__ tvec,
    const float* __restrict__ wrow, float* __restrict__ Y,
    int Np, int K, int M, int act) {
    __shared__ float sB[256 * 64];             // K(<=256) x 64 panel of W

    const int tid     = threadIdx.x;
    const int colBase = blockIdx.y * 64;

    // Cooperative, coalesced panel load: sB[k][c] = W[k][colBase + c].
    // Must run before any wave exits (all waves feed the barrier & the panel).
    for (int i = tid; i < K * 64; i += 256) {
        sB[i] = W[(size_t)(i >> 6) * M + colBase + (i & 63)];
    }
    __syncthreads();

    const int lane    = tid & 31;
    const int wave    = tid >> 5;
    const int rowBase = (blockIdx.x * 8 + wave) * 16;
    if (rowBase >= Np) return;                 // wave-uniform: EXEC all-1 for WMMA
    const int r       = lane & 15;
    const int koff    = (lane < 16) ? 0 : 2;   // ISA 32-bit A 16x4 layout
    const int myRow   = rowBase + r;
    const int ldRow   = (myRow < Np) ? myRow : (Np - 1);  // clamp: unstored D rows

    v8f acc[4];
    #pragma unroll
    for (int j = 0; j < 4; ++j) acc[j] = (v8f){0.f,0.f,0.f,0.f,0.f,0.f,0.f,0.f};

    const float* xp = X + (size_t)ldRow * K + koff;   // A: row=ldRow, K = k+koff, +1
    const float* sp = sB + koff * 64 + r;             // B rows k+koff / k+koff+1 in LDS

    for (int k = 0; k < K; k += 4) {
        v2f a;
        a.x = xp[0];
        a.y = xp[1];
        v2f b0, b1, b2, b3;
        b0.x = sp[0];   b0.y = sp[64];
        b1.x = sp[16];  b1.y = sp[80];
        b2.x = sp[32];  b2.y = sp[96];
        b3.x = sp[48];  b3.y = sp[112];
        acc[0] = __builtin_amdgcn_wmma_f32_16x16x4_f32(false, a, false, b0, (short)0, acc[0], false, false);
        acc[1] = __builtin_amdgcn_wmma_f32_16x16x4_f32(false, a, false, b1, (short)0, acc[1], false, false);
        acc[2] = __builtin_amdgcn_wmma_f32_16x16x4_f32(false, a, false, b2, (short)0, acc[2], false, false);
        acc[3] = __builtin_amdgcn_wmma_f32_16x16x4_f32(false, a, false, b3, (short)0, acc[3], false, false);
        xp += 4;
        sp += 256;                                   // 4 rows * 64 cols
    }

    // C/D layout: VGPR rr -> M = rr + (lane>=16 ? 8 : 0), N = lane&15
    const int rowOff = (lane < 16) ? 0 : 8;
    #pragma unroll
    for (int j = 0; j < 4; ++j) {
        int c = colBase + j * 16 + r;
        #pragma unroll
        for (int rr = 0; rr < 8; ++rr) {
            int row = rowBase + rr + rowOff;
            if (row >= Np) continue;
            float v = acc[j][rr];
            if (tvec) v += tvec[row] * wrow[c];   // layer-1 rank-1 term (t column)
            if (bias) v += bias[c];
            if (act)  v = gelu_tanh(v);
            Y[(size_t)row * M + c] = v;
        }
    }
}

// agg[n,m] = dinv[n]^2 * h[n,m]   (self-loop contribution; norm_self = dinv*dinv)
__global__ void k_selfinit(const float* __restrict__ h, const float* __restrict__ dinv,
                           float* __restrict__ agg, int Np, int M) {
    int i = blockIdx.x * blockDim.x + threadIdx.x;
    if (i < Np * M) {
        int n = i / M;
        float dv = dinv[n];
        agg[i] = dv * dv * h[i];
    }
}

// agg[dst[e], m] += norm[e] * h[src[e], m]  -- one wave per edge
__global__ __launch_bounds__(256) void k_edge_agg(
    const long long* __restrict__ src, const long long* __restrict__ dst,
    const float* __restrict__ norm, const float* __restrict__ h,
    float* __restrict__ agg, int E, int M) {
    int e = blockIdx.x * 8 + (threadIdx.x >> 5);
    if (e >= E) return;
    int lane = threadIdx.x & 31;
    int s = (int)src[e];
    int d = (int)dst[e];
    float nv = norm[e];
    const float* hp = h + (size_t)s * M;
    float* ap = agg + (size_t)d * M;
    for (int m = lane; m < M; m += 32)
        atomicAdd(&ap[m], nv * hp[m]);
}

__global__ void k_bias_gelu(float* __restrict__ p, const float* __restrict__ bias, int Np, int M) {
    int i = blockIdx.x * blockDim.x + threadIdx.x;
    if (i < Np * M) {
        int c = i % M;
        p[i] = gelu_tanh(p[i] + bias[c]);
    }
}

extern "C" void kernel_launch(void* const* d_in, const int* in_sizes, int n_in,
                              void* d_out, int out_size, void* d_ws, size_t ws_size,
                              hipStream_t stream) {
    const float*     x  = (const float*)d_in[0];
    const float*     t  = (const float*)d_in[1];
    const long long* ei = (const long long*)d_in[2];   // int64 edge_index [2,E]
    const float* W1 = (const float*)d_in[3];  const float* b1 = (const float*)d_in[4];
    const float* W2 = (const float*)d_in[5];  const float* b2 = (const float*)d_in[6];
    const float* W3 = (const float*)d_in[7];  const float* b3 = (const float*)d_in[8];
    const float* W4 = (const float*)d_in[9];  const float* b4 = (const float*)d_in[10];
    const float* F1 = (const float*)d_in[11]; const float* c1 = (const float*)d_in[12];
    const float* F2 = (const float*)d_in[13]; const float* c2 = (const float*)d_in[14];
    const float* F3 = (const float*)d_in[15]; const float* c3 = (const float*)d_in[16];

    const int N = in_sizes[1];          // t has N elements
    const int E = in_sizes[2] / 2;
    const long long* src = ei;
    const long long* dst = ei + E;

    // workspace carve
    char* ws = (char*)d_ws;
    size_t off = 0;
    auto carve = [&](size_t bytes) -> void* {
        void* p = ws + off;
        off += (bytes + 255) & ~(size_t)255;
        return p;
    };
    float* dinv = (float*)carve((size_t)N * 4);
    float* norm = (float*)carve((size_t)E * 4);
    float* bufA = (float*)carve((size_t)N * 256 * 4);  // linear output (h_lin)
    float* bufB = (float*)carve((size_t)N * 256 * 4);  // aggregated activations

    // GCN normalization
    k_fill1<<<(N + 255) / 256, 256, 0, stream>>>(dinv, N);
    k_deg  <<<(E + 255) / 256, 256, 0, stream>>>(dst, dinv, E);
    k_dinv <<<(N + 255) / 256, 256, 0, stream>>>(dinv, N);
    k_norm <<<(E + 255) / 256, 256, 0, stream>>>(src, dst, dinv, norm, E);

    const int rowBlocks = (((N + 15) / 16) + 7) / 8;
    auto linear = [&](const float* X, const float* W, const float* bias,
                      const float* tv, const float* wr, float* Y, int K, int M, int act) {
        dim3 grid(rowBlocks, M / 64);
        k_linear_wmma<<<grid, 256, 0, stream>>>(X, W, bias, tv, wr, Y, N, K, M, act);
    };
    auto gcn = [&](const float* Xin, const float* W, const float* bias,
                   const float* tv, const float* wr, int K, int M) {
        linear(Xin, W, nullptr, tv, wr, bufA, K, M, 0);
        int tot = N * M;
        k_selfinit <<<(tot + 255) / 256, 256, 0, stream>>>(bufA, dinv, bufB, N, M);
        k_edge_agg <<<(E + 7) / 8, 256, 0, stream>>>(src, dst, norm, bufA, bufB, E, M);
        k_bias_gelu<<<(tot + 255) / 256, 256, 0, stream>>>(bufB, bias, N, M);
    };

    // 4 GCN layers (layer 1: [x | t] @ W1 via rank-1 epilogue with W1 row 64)
    gcn(x,    W1, b1, t,       W1 + 64 * 64, 64, 64);
    gcn(bufB, W2, b2, nullptr, nullptr,      64, 128);
    gcn(bufB, W3, b3, nullptr, nullptr,     128, 256);
    gcn(bufB, W4, b4, nullptr, nullptr,     256, 128);

    // MLP head
    linear(bufB, F1, c1, nullptr, nullptr, bufA,          128, 256, 1);
    linear(bufA, F2, c2, nullptr, nullptr, bufB,          256, 128, 1);
    linear(bufB, F3, c3, nullptr, nullptr, (float*)d_out, 128,  64, 0);
}